// GeneralAffinityCalculator_84499186582124
// MI455X (gfx1250) — compile-verified
//
#include <hip/hip_runtime.h>
#include <hip/hip_bf16.h>
#include <stdint.h>

// Problem constants (match reference)
#define LATENT 128
#define DPROJ  32
#define BB     8
#define NN     16384
#define KK     16

typedef __attribute__((ext_vector_type(2))) float v2f;
typedef __attribute__((ext_vector_type(8))) float v8f;

// ---------------------------------------------------------------------------
// Phase 1: P = F @ W^T + b for ks and qs, via V_WMMA_F32_16X16X4_F32.
// One wave computes 16 rows x 32 cols for BOTH projections (4 f32 16x16 tiles).
// Block = 256 threads = 8 waves -> 128 rows per block.
// ---------------------------------------------------------------------------
__global__ __launch_bounds__(256)
void proj_wmma_kernel(const float* __restrict__ F,
                      const float* __restrict__ ks_w, const float* __restrict__ ks_b,
                      const float* __restrict__ qs_w, const float* __restrict__ qs_b,
                      float* __restrict__ Pk, float* __restrict__ Pq)
{
    const int tid  = threadIdx.x;
    const int wave = tid >> 5;
    const int lane = tid & 31;
    const int half = lane >> 4;   // ISA A/B layout: lanes 16-31 hold K+2,K+3
    const int l15  = lane & 15;
    const int koff = half * 2;

    const long rowbase = (long)blockIdx.x * 128 + (long)wave * 16;

    // A fragment base: row = rowbase + (lane&15), this lane's K pair at +koff
    const float* aptr   = F    + (rowbase + l15) * LATENT + koff;
    // B fragments: B[k][n] = W[n][k]; lane owns column n = l15 (+16 for hi tile)
    const float* bks_lo = ks_w + (long)l15 * LATENT + koff;
    const float* bks_hi = bks_lo + 16 * LATENT;
    const float* bqs_lo = qs_w + (long)l15 * LATENT + koff;
    const float* bqs_hi = bqs_lo + 16 * LATENT;

    v8f acc_k0 = {}, acc_k1 = {}, acc_q0 = {}, acc_q1 = {};

#pragma unroll 4
    for (int k = 0; k < LATENT; k += 4) {
        v2f a   = *(const v2f*)(aptr   + k);
        v2f bk0 = *(const v2f*)(bks_lo + k);
        v2f bk1 = *(const v2f*)(bks_hi + k);
        v2f bq0 = *(const v2f*)(bqs_lo + k);
        v2f bq1 = *(const v2f*)(bqs_hi + k);
        // 8-arg form: (neg_a, A, neg_b, B, c_mod, C, reuse_a, reuse_b)
        acc_k0 = __builtin_amdgcn_wmma_f32_16x16x4_f32(false, a, false, bk0, (short)0, acc_k0, false, false);
        acc_k1 = __builtin_amdgcn_wmma_f32_16x16x4_f32(false, a, false, bk1, (short)0, acc_k1, false, false);
        acc_q0 = __builtin_amdgcn_wmma_f32_16x16x4_f32(false, a, false, bq0, (short)0, acc_q0, false, false);
        acc_q1 = __builtin_amdgcn_wmma_f32_16x16x4_f32(false, a, false, bq1, (short)0, acc_q1, false, false);
    }

    // Epilogue. C/D layout: VGPR r, lanes 0-15 -> M=r, lanes 16-31 -> M=r+8; N=lane&15.
    const float bk_lo = ks_b[l15], bk_hi = ks_b[l15 + 16];
    const float bq_lo = qs_b[l15], bq_hi = qs_b[l15 + 16];
#pragma unroll
    for (int r = 0; r < 8; ++r) {
        long row = rowbase + r + 8 * half;
        float* pk = Pk + row * DPROJ + l15;
        float* pq = Pq + row * DPROJ + l15;
        pk[0]  = acc_k0[r] + bk_lo;
        pk[16] = acc_k1[r] + bk_hi;
        pq[0]  = acc_q0[r] + bq_lo;
        pq[16] = acc_q1[r] + bq_hi;
    }
}

// ---------------------------------------------------------------------------
// Phase 2: gather + scaled dot. 8 lanes cooperate on one output:
// each lane loads one float4 (16B) of the 128B row -> a single coalesced 128B
// transaction per gathered row, then 3-step shfl_xor tree reduction.
// ---------------------------------------------------------------------------
__global__ __launch_bounds__(256)
void affinity_gather_kernel(const float* __restrict__ Pk,
                            const float* __restrict__ Pq,
                            const long long* __restrict__ idx,
                            float* __restrict__ out)
{
    const long long tid = (long long)blockIdx.x * 256 + threadIdx.x;
    const long long o   = tid >> 3;   // output element index in [0, B*N*K)
    const int       c   = (int)(tid & 7);

    const long long NK  = (long long)NN * KK;
    const int       b   = (int)(o / NK);
    const long long rem = o - (long long)b * NK;     // n*K + k

    const long long xi = idx[((long long)(1 * BB + b)) * NK + rem];
    const long long yi = idx[((long long)(2 * BB + b)) * NK + rem];

    const float4 x = *(const float4*)(Pk + ((long long)b * NN + xi) * DPROJ + c * 4);
    const float4 y = *(const float4*)(Pq + ((long long)b * NN + yi) * DPROJ + c * 4);

    float p = x.x * y.x + x.y * y.y + x.z * y.z + x.w * y.w;
    p += __shfl_xor(p, 1, 32);
    p += __shfl_xor(p, 2, 32);
    p += __shfl_xor(p, 4, 32);

    if (c == 0) out[o] = p * 0.17677669529663689f;  // 32^-0.5
}

// ---------------------------------------------------------------------------
extern "C" void kernel_launch(void* const* d_in, const int* in_sizes, int n_in,
                              void* d_out, int out_size, void* d_ws, size_t ws_size,
                              hipStream_t stream) {
    const float*     F    = (const float*)d_in[0];
    const float*     ks_w = (const float*)d_in[1];
    const float*     ks_b = (const float*)d_in[2];
    const float*     qs_w = (const float*)d_in[3];
    const float*     qs_b = (const float*)d_in[4];
    const long long* idx  = (const long long*)d_in[5];
    float*           out  = (float*)d_out;

    float* Pk = (float*)d_ws;                               // [B*N, 32]
    float* Pq = Pk + (size_t)BB * NN * DPROJ;               // [B*N, 32]

    // Phase 1: (B*N)/128 = 1024 blocks of 256 threads.
    proj_wmma_kernel<<<(BB * NN) / 128, 256, 0, stream>>>(F, ks_w, ks_b, qs_w, qs_b, Pk, Pq);

    // Phase 2: B*N*K outputs * 8 lanes each / 256 threads = 65536 blocks.
    affinity_gather_kernel<<<(BB * NN * KK * 8) / 256, 256, 0, stream>>>(Pk, Pq, idx, out);
}